// AutoCorrelationFast_90469191123603
// MI455X (gfx1250) — compile-verified
//
#include <hip/hip_runtime.h>
#include <hip/hip_bf16.h>
#include <math.h>

// Problem constants (from reference)
constexpr int D      = 1024;   // d_model
constexpr int LSEQ   = 4096;   // sequence length
constexpr int BB     = 2;      // batch
constexpr int HH     = 16;     // heads
constexpr int DKH    = 64;     // head dim
constexpr int MROWS  = BB * LSEQ;  // 8192
constexpr int TOPK   = 8;          // int(log(4097)) == 8

typedef float v2f __attribute__((ext_vector_type(2)));
typedef float v4f __attribute__((ext_vector_type(4)));
typedef float v8f __attribute__((ext_vector_type(8)));

// ---------------------------------------------------------------------------
// GEMM: out = X[M=8192,K=1024] @ W[1024,1024] + bias, f32 WMMA 16x16x4.
// Register-blocked: workgroup (128 thr / 4 waves) computes a 64x64 tile.
// Wave w owns rows [w*16, w*16+16) x all 64 cols -> 4 v8f accumulators.
// Per kk-step: 1 A frag + 4 B frags -> 4 WMMAs (A reuse x4; B shared by all
// waves via LDS). A (64x32) staged float4-coalesced; W (32x64) staged
// transposed so B frags are contiguous 8B LDS reads.
// headMajor=1 stores out[b,h,l,d] (head-major) for the attention-side kernels.
// ---------------------------------------------------------------------------
__global__ __launch_bounds__(128) void gemm_wmma_f32(
    const float* __restrict__ X, const float* __restrict__ W,
    const float* __restrict__ bias, float* __restrict__ out, int headMajor)
{
  __shared__ __align__(16) float aS[64 * 32];   // [row][k]
  __shared__ __align__(16) float bS[64 * 32];   // transposed: [n][k]
  const int tid  = threadIdx.x;
  const int wave = tid >> 5;
  const int lane = tid & 31;
  const int m0 = blockIdx.y * 64;
  const int n0 = blockIdx.x * 64;
  const int hi = lane >> 4;      // which K-pair half (K=2,3 on high lanes)
  const int ln = lane & 15;

  v8f acc[4] = {};

  for (int k0 = 0; k0 < D; k0 += 32) {
    // Stage A tile 64x32 with float4 (rows of 32 floats, coalesced)
    for (int i = tid * 4; i < 64 * 32; i += 128 * 4) {
      int r = i >> 5, c = i & 31;
      *(v4f*)&aS[i] = *(const v4f*)&X[(size_t)(m0 + r) * D + k0 + c];
    }
    // Stage W tile 32x64 transposed into bS[n][k] (coalesced global reads)
    for (int i = tid; i < 32 * 64; i += 128) {
      int k = i >> 6, n = i & 63;
      bS[n * 32 + k] = W[(size_t)(k0 + k) * D + n0 + n];
    }
    if (k0 + 32 < D) {  // global_prefetch_b8: pull next tiles toward L0/L2
      __builtin_prefetch(&X[(size_t)(m0 + (tid >> 1)) * D + k0 + 32 + (tid & 1) * 16], 0, 1);
      __builtin_prefetch(&W[(size_t)(k0 + 32 + (tid >> 2)) * D + n0 + (tid & 3) * 16], 0, 1);
    }
    __syncthreads();

    const float* ap = &aS[(wave * 16 + ln) * 32 + hi * 2];
    const float* bp = &bS[ln * 32 + hi * 2];
#pragma unroll
    for (int kk = 0; kk < 32; kk += 4) {
      v2f a = *(const v2f*)(ap + kk);   // A[m][kb], A[m][kb+1]
#pragma unroll
      for (int nt = 0; nt < 4; ++nt) {
        v2f b = *(const v2f*)(bp + nt * 16 * 32 + kk);  // B[kb][n], B[kb+1][n]
        acc[nt] = __builtin_amdgcn_wmma_f32_16x16x4_f32(
            false, a, false, b, (short)0, acc[nt], false, false);
      }
    }
    __syncthreads();
  }

#pragma unroll
  for (int nt = 0; nt < 4; ++nt) {
    const int col = n0 + nt * 16 + ln;
    const float bv = bias[col];
#pragma unroll
    for (int r = 0; r < 8; ++r) {
      int row = m0 + wave * 16 + r + 8 * hi;   // C/D layout: M = r + 8*hi
      float v = acc[nt][r] + bv;
      if (headMajor) {
        int b = row >> 12, l = row & (LSEQ - 1);
        int h = col >> 6,  d = col & (DKH - 1);
        out[(((size_t)(b * HH + h) * LSEQ + l) << 6) + d] = v;
      } else {
        out[(size_t)row * D + col] = v;
      }
    }
  }
}

// ---------------------------------------------------------------------------
// Autocorrelation via banded Q*K^T: corr[bh][tau] = (1/64)*sum_l Q[l].K[(l-tau)%L]
// One wave per (head, diag-band). Band D covers all tile pairs with
// (l_tile - m_tile) mod 256 == D; every element of such a tile maps to
// tau = (16*D + i - j) mod 4096, constant across the band -> the WMMA
// accumulator sums the whole band; one diagonal reduction at the end.
// ---------------------------------------------------------------------------
__global__ __launch_bounds__(32) void corr_band_wmma(
    const float* __restrict__ Qh, const float* __restrict__ Kh,
    float* __restrict__ corr)
{
  __shared__ __align__(16) float qS[16 * DKH];
  __shared__ __align__(16) float kS[16 * DKH];
  __shared__ float bins[31];
  const int lane = threadIdx.x;
  const int band = blockIdx.x;   // 0..255
  const int bh   = blockIdx.y;   // 0..31
  const float* Qb = Qh + (size_t)bh * LSEQ * DKH;
  const float* Kb = Kh + (size_t)bh * LSEQ * DKH;
  const int hi = lane >> 4, ln = lane & 15;

  v8f acc = {};
  for (int t = 0; t < 256; ++t) {
    const int mt = (t - band) & 255;
    const float* qsrc = Qb + t  * 16 * DKH;
    const float* ksrc = Kb + mt * 16 * DKH;
    for (int i = lane * 4; i < 16 * DKH; i += 128) {  // float4-coalesced stage
      *(v4f*)&qS[i] = *(const v4f*)&qsrc[i];
      *(v4f*)&kS[i] = *(const v4f*)&ksrc[i];
    }
    if (t + 1 < 256) {   // prefetch next 4KB Q/K tiles
      __builtin_prefetch(qsrc + 16 * DKH + lane * 32, 0, 1);
      __builtin_prefetch(Kb + (((t + 1 - band) & 255) * 16 * DKH) + lane * 32, 0, 1);
    }
    __syncthreads();
    const float* ap = &qS[ln * DKH + hi * 2];
    const float* bp = &kS[ln * DKH + hi * 2];   // B[k][n] = K[n][k] (contiguous)
#pragma unroll
    for (int kk = 0; kk < DKH; kk += 4) {
      v2f a = *(const v2f*)(ap + kk);
      v2f b = *(const v2f*)(bp + kk);
      acc = __builtin_amdgcn_wmma_f32_16x16x4_f32(
          false, a, false, b, (short)0, acc, false, false);
    }
    __syncthreads();
  }

  if (lane < 31) bins[lane] = 0.0f;
  __syncthreads();
#pragma unroll
  for (int r = 0; r < 8; ++r) {
    int M = r + 8 * hi;           // i (row in l)
    int N = ln;                   // j (row in m)
    atomicAdd(&bins[M - N + 15], acc[r]);   // ds_add_f32
  }
  __syncthreads();
  if (lane < 31) {
    int delta = lane - 15;
    int tau = (band * 16 + delta) & (LSEQ - 1);
    atomicAdd(&corr[bh * LSEQ + tau], bins[lane] * (1.0f / 64.0f));
  }
}

// ---------------------------------------------------------------------------
// Per-(b,h) top-8 (lowest-index tie-break, matching lax.top_k) + softmax.
// corr row lives entirely in LDS; 8 masked argmax rounds.
// ---------------------------------------------------------------------------
__global__ __launch_bounds__(256) void topk_softmax(
    const float* __restrict__ corr, float* __restrict__ wOut,
    int* __restrict__ iOut)
{
  __shared__ float vals[LSEQ];
  __shared__ float rv[256];
  __shared__ int   ri[256];
  __shared__ float topv[TOPK];
  __shared__ int   topi[TOPK];
  const int tid = threadIdx.x;
  const int bh  = blockIdx.x;
  for (int i = tid; i < LSEQ; i += 256) vals[i] = corr[bh * LSEQ + i];
  __syncthreads();
  for (int r = 0; r < TOPK; ++r) {
    float bestv = -INFINITY; int besti = 0;
    for (int i = tid; i < LSEQ; i += 256) {
      float v = vals[i];
      if (v > bestv) { bestv = v; besti = i; }  // first hit = lowest index
    }
    rv[tid] = bestv; ri[tid] = besti;
    __syncthreads();
    for (int s = 128; s > 0; s >>= 1) {
      if (tid < s) {
        float v2 = rv[tid + s]; int i2 = ri[tid + s];
        if (v2 > rv[tid] || (v2 == rv[tid] && i2 < ri[tid])) {
          rv[tid] = v2; ri[tid] = i2;
        }
      }
      __syncthreads();
    }
    if (tid == 0) { topv[r] = rv[0]; topi[r] = ri[0]; vals[ri[0]] = -INFINITY; }
    __syncthreads();
  }
  if (tid == 0) {
    float m = topv[0];                      // sorted descending
    float e[TOPK], s = 0.0f;
    for (int k = 0; k < TOPK; ++k) { e[k] = __expf(topv[k] - m); s += e[k]; }
    float inv = 1.0f / s;
    for (int k = 0; k < TOPK; ++k) {
      wOut[bh * TOPK + k] = e[k] * inv;
      iOut[bh * TOPK + k] = topi[k];
    }
  }
}

// ---------------------------------------------------------------------------
// out[b,l,h*64+d] = sum_k w[bh,k] * V[b,h,(l+idx[bh,k])%L, d]  (float4 lanes)
// ---------------------------------------------------------------------------
__global__ __launch_bounds__(256) void gather_roll(
    const float* __restrict__ Vh, const float* __restrict__ wTab,
    const int* __restrict__ iTab, float* __restrict__ X2)
{
  int g = blockIdx.x * 256 + threadIdx.x;       // float4 id, 2,097,152 total
  int col4 = g & 255;                           // 1024/4 per row
  int row  = g >> 8;                            // b*4096 + l
  int h = col4 >> 4;
  int d = (col4 & 15) << 2;
  int b = row >> 12;
  int l = row & (LSEQ - 1);
  int bh = b * HH + h;
  const float* Vb = Vh + (size_t)bh * LSEQ * DKH;
  v4f acc = {};
#pragma unroll
  for (int k = 0; k < TOPK; ++k) {
    float w  = wTab[bh * TOPK + k];
    int   ls = (l + iTab[bh * TOPK + k]) & (LSEQ - 1);
    v4f v = *(const v4f*)&Vb[ls * DKH + d];
    acc += v * w;
  }
  *(v4f*)&X2[(size_t)row * D + col4 * 4] = acc;
}

__global__ void zero_f32(float* __restrict__ p, int n) {
  int i = blockIdx.x * 256 + threadIdx.x;
  if (i < n) p[i] = 0.0f;
}

// ---------------------------------------------------------------------------
extern "C" void kernel_launch(void* const* d_in, const int* in_sizes, int n_in,
                              void* d_out, int out_size, void* d_ws, size_t ws_size,
                              hipStream_t stream) {
  const float* query = (const float*)d_in[0];
  const float* key   = (const float*)d_in[1];
  const float* value = (const float*)d_in[2];
  const float* Wq    = (const float*)d_in[3];
  const float* bq    = (const float*)d_in[4];
  const float* Wk    = (const float*)d_in[5];
  const float* bk    = (const float*)d_in[6];
  const float* Wv    = (const float*)d_in[7];
  const float* bv    = (const float*)d_in[8];
  const float* Wo    = (const float*)d_in[9];
  const float* bo    = (const float*)d_in[10];

  // Workspace layout:
  //   Qh 33.5MB | Kh 33.5MB | Vh 33.5MB | corr 512KB | w 1KB | idx 1KB
  //   X2 reuses the Qh region (Qh dead after corr_band). Total ~101.2 MB.
  float* Qh   = (float*)d_ws;
  float* Kh   = Qh + (size_t)MROWS * D;
  float* Vh   = Kh + (size_t)MROWS * D;
  float* corr = Vh + (size_t)MROWS * D;          // 32*4096 floats
  float* wTab = corr + BB * HH * LSEQ;
  int*   iTab = (int*)(wTab + BB * HH * TOPK);
  float* X2   = Qh;

  dim3 gGemm(D / 64, MROWS / 64), bGemm(128);
  gemm_wmma_f32<<<gGemm, bGemm, 0, stream>>>(query, Wq, bq, Qh, 1);
  gemm_wmma_f32<<<gGemm, bGemm, 0, stream>>>(key,   Wk, bk, Kh, 1);
  gemm_wmma_f32<<<gGemm, bGemm, 0, stream>>>(value, Wv, bv, Vh, 1);

  int corrN = BB * HH * LSEQ;
  zero_f32<<<(corrN + 255) / 256, 256, 0, stream>>>(corr, corrN);
  corr_band_wmma<<<dim3(256, BB * HH), 32, 0, stream>>>(Qh, Kh, corr);
  topk_softmax<<<BB * HH, 256, 0, stream>>>(corr, wTab, iTab);

  gather_roll<<<(MROWS * D / 4) / 256, 256, 0, stream>>>(Vh, wTab, iTab, X2);
  gemm_wmma_f32<<<gGemm, bGemm, 0, stream>>>(X2, Wo, bo, (float*)d_out, 0);
}